// DynamicSoftKMeansLoss_74594991997729
// MI455X (gfx1250) — compile-verified
//
#include <hip/hip_runtime.h>
#include <math.h>

typedef __attribute__((ext_vector_type(2))) float v2f;
typedef __attribute__((ext_vector_type(8))) float v8f;

#define ROWS 260          // 256 + 4 pad floats -> conflict-free b64 LDS fragment reads
#define MAXC 16
#define MAXK 16
// acc layout (floats): 0:cnt[16] 16:wd2[16] 32:viol[16] 48:present[16] 64:sdist[16*16]
// 320:n_unique 321..336: closest[] (stored as int bits)
#define ACC_CNT   0
#define ACC_WD2   16
#define ACC_VIOL  32
#define ACC_PRES  48
#define ACC_SDIST 64
#define ACC_NUNIQ 320
#define ACC_CLOSEST 321
#define ACC_FLOATS 352

__global__ void kzero(float* acc) {
    int i = blockIdx.x * blockDim.x + threadIdx.x;
    if (i < ACC_FLOATS) acc[i] = 0.0f;
}

// ---------------- Main pass: WMMA f32 16x16x4, one wave (32 threads) per block,
// double-buffered async global->LDS staging (ASYNCcnt) ----------------
__global__ void kmain(const float* __restrict__ feat,
                      const int* __restrict__ labels,
                      const int* __restrict__ label2,
                      const float* __restrict__ centers,
                      float* __restrict__ wsDist,
                      float* __restrict__ wsWd,
                      float* __restrict__ acc,
                      int tiles, int K) {
    __shared__ __align__(16) float sTile[2][16 * ROWS]; // double-buffered feature tiles
    __shared__ __align__(16) float sCent[16 * ROWS];    // centers zero-padded to 16 rows
    __shared__ float sDots[16 * 16];
    __shared__ float sCC[16];
    __shared__ float sCnt[16], sWd2[16], sPres[16];
    __shared__ float sSd[16 * 16];

    const int lane = threadIdx.x;          // blockDim.x == 32, one wave
    const int half = lane >> 4;            // A/B fragment half-wave select
    const int lm   = lane & 15;

    // async-stage a 16x256 feature tile into buffer `buf` (32 b128 loads / lane)
    auto issueTile = [&](int t, int buf) {
        unsigned ldsBase = (unsigned)(size_t)(void*)(&sTile[buf][0]);
        unsigned gbase = (unsigned)((size_t)t * 16 * 256 * 4);   // byte offset, < 2^31 here
        #pragma unroll 8
        for (int it = 0; it < 32; ++it) {
            int idx = lane + 32 * it;            // 0..1023 : 16 rows x 64 float4
            int row = idx >> 6, c4 = idx & 63;
            unsigned lds  = ldsBase + (unsigned)((row * ROWS + c4 * 4) * 4);
            unsigned goff = gbase + (unsigned)((row * 256 + c4 * 4) * 4);
            asm volatile("global_load_async_to_lds_b128 %0, %1, %2"
                         :: "v"(lds), "v"(goff), "s"(feat)
                         : "memory");
        }
    };

    // prefetch very first tile before doing anything else
    if (blockIdx.x < (unsigned)tiles) issueTile(blockIdx.x, 0);

    // stage centers once with regular loads (rows >= K are zero -> inert WMMA columns)
    #pragma unroll 4
    for (int it = 0; it < 32; ++it) {
        int idx = lane + 32 * it;
        int row = idx >> 6, c4 = idx & 63;
        float4 v = make_float4(0.f, 0.f, 0.f, 0.f);
        if (row < K) v = *(const float4*)(centers + row * 256 + c4 * 4);
        *(float4*)(sCent + row * ROWS + c4 * 4) = v;
    }
    // zero LDS accumulators
    for (int i = lane; i < 16 * 16; i += 32) sSd[i] = 0.0f;
    if (lane < 16) { sCnt[lane] = 0.0f; sWd2[lane] = 0.0f; sPres[lane] = 0.0f; }
    __syncthreads();
    // per-center squared norms
    if (lane < 16) {
        float s = 0.0f;
        const float4* cp = (const float4*)(sCent + lane * ROWS);
        for (int c4 = 0; c4 < 64; ++c4) {
            float4 v = cp[c4];
            s += v.x * v.x + v.y * v.y + v.z * v.z + v.w * v.w;
        }
        sCC[lane] = s;
    }
    __syncthreads();

    int buf = 0;
    for (int t = blockIdx.x; t < tiles; t += gridDim.x) {
        // current buffer's async loads must have landed
        asm volatile("s_wait_asynccnt 0" ::: "memory");

        // kick off next tile into the other buffer (its last readers were fenced
        // by the previous iteration's __syncthreads -> s_wait_dscnt 0)
        int tn = t + gridDim.x;
        if (tn < tiles) issueTile(tn, buf ^ 1);

        const float* tp = &sTile[buf][0];

        // ---- D[16x16] = featTile[16x256] x centersT[256x16] via 64 k-steps of 4 ----
        v8f accA = {0.f,0.f,0.f,0.f,0.f,0.f,0.f,0.f};
        v8f accB = {0.f,0.f,0.f,0.f,0.f,0.f,0.f,0.f};
        const float* aBase = tp    + lm * ROWS + 2 * half;  // A: lane holds row lm
        const float* bBase = sCent + lm * ROWS + 2 * half;  // B: lane holds col lm (center)
        #pragma unroll 8
        for (int kk = 0; kk < 64; kk += 2) {
            v2f a0 = *(const v2f*)(aBase + kk * 4);
            v2f b0 = *(const v2f*)(bBase + kk * 4);
            accA = __builtin_amdgcn_wmma_f32_16x16x4_f32(false, a0, false, b0,
                                                         (short)0, accA, false, false);
            v2f a1 = *(const v2f*)(aBase + kk * 4 + 4);
            v2f b1 = *(const v2f*)(bBase + kk * 4 + 4);
            accB = __builtin_amdgcn_wmma_f32_16x16x4_f32(false, a1, false, b1,
                                                         (short)0, accB, false, false);
        }
        v8f accD = accA + accB;

        // scatter dots to LDS: VGPR v holds M = v + 8*half, N = lm
        if (lm < K) {
            #pragma unroll
            for (int v = 0; v < 8; ++v)
                sDots[(v + 8 * half) * 16 + lm] = accD[v];
        }
        __syncthreads();

        // ---- per-row epilogue: 16 lanes, one row each ----
        if (lane < 16) {
            const int row = lane;
            const size_t i = (size_t)t * 16 + row;
            float xx = 0.0f;
            const float4* rp = (const float4*)(tp + row * ROWS);
            for (int c4 = 0; c4 < 64; ++c4) {
                float4 v = rp[c4];
                xx += v.x * v.x + v.y * v.y + v.z * v.z + v.w * v.w;
            }
            float dist[MAXK];
            float dmin = 3.4e38f;
            for (int n = 0; n < K; ++n) {
                float d2 = xx + sCC[n] - 2.0f * sDots[row * 16 + n];
                float di = sqrtf(fmaxf(d2, 1e-12f));
                dist[n] = di;
                dmin = fminf(dmin, di);
            }
            // softmax(-dist), weighted distance
            float s = 0.0f, wds = 0.0f;
            for (int n = 0; n < K; ++n) {
                float e = __expf(dmin - dist[n]);
                s += e;
                wds += e * dist[n];
            }
            float wd = wds / s;

            int lab = labels[i] & (MAXC - 1);
            float w = (label2[i] == 1) ? 1.0f : 0.0f;

            wsWd[i] = wd;
            for (int n = 0; n < K; ++n) wsDist[i * K + n] = dist[n];

            atomicAdd(&sCnt[lab], w);
            atomicAdd(&sWd2[lab], w * wd * wd);
            atomicAdd(&sPres[lab], 1.0f);
            for (int n = 0; n < K; ++n) atomicAdd(&sSd[lab * 16 + n], w * dist[n]);
        }
        __syncthreads();   // fences DS reads on this buffer before it is refilled
        buf ^= 1;
    }

    // ---- flush block-local sums once ----
    if (lane < 16) {
        atomicAdd(&acc[ACC_CNT  + lane], sCnt[lane]);
        atomicAdd(&acc[ACC_WD2  + lane], sWd2[lane]);
        atomicAdd(&acc[ACC_PRES + lane], sPres[lane]);
    }
    for (int i = lane; i < 16 * 16; i += 32) atomicAdd(&acc[ACC_SDIST + i], sSd[i]);
}

// ---------------- Generic scalar fallback (remainder rows / D != 256) ----------------
__global__ void krow(const float* __restrict__ feat,
                     const int* __restrict__ labels,
                     const int* __restrict__ label2,
                     const float* __restrict__ centers,
                     float* __restrict__ wsDist,
                     float* __restrict__ wsWd,
                     float* __restrict__ acc,
                     int start, int B, int D, int K) {
    int KK = K > MAXK ? MAXK : K;
    for (size_t i = start + (size_t)blockIdx.x * blockDim.x + threadIdx.x; i < (size_t)B;
         i += (size_t)gridDim.x * blockDim.x) {
        const float* x = feat + i * D;
        float xx = 0.0f;
        for (int d = 0; d < D; ++d) xx += x[d] * x[d];
        float dist[MAXK];
        float dmin = 3.4e38f;
        for (int n = 0; n < KK; ++n) {
            const float* c = centers + (size_t)n * D;
            float dot = 0.0f, cc = 0.0f;
            for (int d = 0; d < D; ++d) { dot += x[d] * c[d]; cc += c[d] * c[d]; }
            float d2 = xx + cc - 2.0f * dot;
            float di = sqrtf(fmaxf(d2, 1e-12f));
            dist[n] = di;
            dmin = fminf(dmin, di);
        }
        float s = 0.0f, wds = 0.0f;
        for (int n = 0; n < KK; ++n) {
            float e = __expf(dmin - dist[n]);
            s += e; wds += e * dist[n];
        }
        float wd = wds / s;
        int lab = labels[i] & (MAXC - 1);
        float w = (label2[i] == 1) ? 1.0f : 0.0f;
        wsWd[i] = wd;
        for (int n = 0; n < KK; ++n) wsDist[i * K + n] = dist[n];
        atomicAdd(&acc[ACC_CNT + lab], w);
        atomicAdd(&acc[ACC_WD2 + lab], w * wd * wd);
        atomicAdd(&acc[ACC_PRES + lab], 1.0f);
        for (int n = 0; n < KK; ++n) atomicAdd(&acc[ACC_SDIST + lab * 16 + n], w * dist[n]);
    }
}

// ---------------- Per-class argmin + unique-class count ----------------
__global__ void kstats(float* acc, const int* pC, int K) {
    __shared__ float pres[32];
    int C = *pC; if (C > MAXC) C = MAXC;
    int KK = K > MAXK ? MAXK : K;
    int c = threadIdx.x;
    float flag = 0.0f;
    if (c < C) {
        float cnt = acc[ACC_CNT + c];
        float safe = fmaxf(cnt, 1.0f);
        float best = 3.4e38f;
        int bi = 0;
        for (int n = 0; n < KK; ++n) {
            float v = acc[ACC_SDIST + c * 16 + n] / safe;
            if (v < best) { best = v; bi = n; }   // first-index tie-break like argmin
        }
        ((int*)acc)[ACC_CLOSEST + c] = bi;
        flag = (acc[ACC_PRES + c] > 0.0f) ? 1.0f : 0.0f;
    }
    pres[threadIdx.x] = flag;
    __syncthreads();
    if (threadIdx.x == 0) {
        float s = 0.0f;
        for (int j = 0; j < 32; ++j) s += pres[j];
        acc[ACC_NUNIQ] = fmaxf(s, 1.0f);
    }
}

// ---------------- Second pass over spilled dist/wd (L2-resident, ~5MB) ----------------
__global__ void kviol(const int* __restrict__ labels,
                      const int* __restrict__ label2,
                      const float* __restrict__ wsDist,
                      const float* __restrict__ wsWd,
                      float* __restrict__ acc,
                      int B, int K) {
    __shared__ float sv[16];
    if (threadIdx.x < 16) sv[threadIdx.x] = 0.0f;
    __syncthreads();
    int KK = K > MAXK ? MAXK : K;
    for (size_t i = (size_t)blockIdx.x * blockDim.x + threadIdx.x; i < (size_t)B;
         i += (size_t)gridDim.x * blockDim.x) {
        if (label2[i] == 1) {
            int lab = labels[i] & (MAXC - 1);
            int excl = ((const int*)acc)[ACC_CLOSEST + lab];
            float mn = 3.4e38f;
            for (int n = 0; n < KK; ++n)
                if (n != excl) mn = fminf(mn, wsDist[i * K + n]);
            float viol = wsWd[i] + 0.5f - mn;     // MARGIN = 0.5
            if (viol > 0.0f) atomicAdd(&sv[lab], viol);
        }
    }
    __syncthreads();
    if (threadIdx.x < 16) atomicAdd(&acc[ACC_VIOL + threadIdx.x], sv[threadIdx.x]);
}

__global__ void kfinal(const float* __restrict__ acc, const int* pC, float* out) {
    if (threadIdx.x == 0 && blockIdx.x == 0) {
        int C = *pC; if (C > MAXC) C = MAXC;
        float s = 0.0f;
        for (int c = 0; c < C; ++c) {
            float cnt = acc[ACC_CNT + c];
            float safe = fmaxf(cnt, 1.0f);
            float hm = (cnt > 0.0f) ? 1.0f : 0.0f;
            s += (acc[ACC_WD2 + c] + acc[ACC_VIOL + c]) / safe * hm;
        }
        out[0] = s / acc[ACC_NUNIQ];
    }
}

extern "C" void kernel_launch(void* const* d_in, const int* in_sizes, int n_in,
                              void* d_out, int out_size, void* d_ws, size_t ws_size,
                              hipStream_t stream) {
    const float* feat    = (const float*)d_in[0];
    const int*   labels  = (const int*)d_in[1];
    const int*   label2  = (const int*)d_in[2];
    const int*   pC      = (const int*)d_in[3];   // device scalar
    const float* centers = (const float*)d_in[4];

    const int B = in_sizes[1];
    const int D = in_sizes[0] / B;
    const int K = in_sizes[4] / D;

    float* ws = (float*)d_ws;
    size_t offWd = (size_t)B * K;
    size_t accF  = ((offWd + (size_t)B + 63) / 64) * 64;
    float* wsDist = ws;
    float* wsWd   = ws + offWd;
    float* acc    = ws + accF;

    kzero<<<2, 256, 0, stream>>>(acc);

    const bool fast = (D == 256) && (K <= MAXK);
    const int tiles = fast ? (B / 16) : 0;
    if (tiles > 0) {
        int blocks = tiles < 2560 ? tiles : 2560;
        kmain<<<blocks, 32, 0, stream>>>(feat, labels, label2, centers,
                                         wsDist, wsWd, acc, tiles, K);
    }
    const int start = tiles * 16;
    if (start < B) {
        int rows = B - start;
        int blocks = (rows + 255) / 256;
        if (blocks > 2048) blocks = 2048;
        krow<<<blocks, 256, 0, stream>>>(feat, labels, label2, centers,
                                         wsDist, wsWd, acc, start, B, D, K);
    }
    kstats<<<1, 32, 0, stream>>>(acc, pC, K);
    {
        int blocks = (B + 255) / 256;
        if (blocks > 2048) blocks = 2048;
        kviol<<<blocks, 256, 0, stream>>>(labels, label2, wsDist, wsWd, acc, B, K);
    }
    kfinal<<<1, 32, 0, stream>>>(acc, pC, (float*)d_out);
}